// DecoderRNN_42425686950052
// MI455X (gfx1250) — compile-verified
//
#include <hip/hip_runtime.h>

// ---------------------------------------------------------------------------
// DecoderRNN single step for MI455X (gfx1250, wave32).
//   out[0..V)   = log_softmax(h_new @ W_out^T + b_out)
//   out[V..V+H) = h_new
// Bandwidth-bound problem (W_out stream = 206MB); the vocab GEMV runs on the
// fp32 WMMA pipe (v_wmma_f32_16x16x4_f32) with a K-permutation that makes
// every A fragment fall directly out of a coalescible per-lane b128 load.
// ---------------------------------------------------------------------------

#define Hdim 1024
#define Vdim 50257
#define H3   (3 * Hdim)
#define NGROUPS ((Vdim + 15) / 16)          // 3142 groups of 16 vocab rows

typedef __attribute__((ext_vector_type(2))) float v2f;
typedef __attribute__((ext_vector_type(8))) float v8f;

// ---------------------------------------------------------------------------
// Kernel 1: gi[r] = W_ih[r,:]·relu(emb[tok]) + b_ih[r]   (r in [0,3H))
//           gh[r] = W_hh[r,:]·h               + b_hh[r]
// One wave per output row; 8 x float4 per lane, shuffle reduction.
// ---------------------------------------------------------------------------
__global__ void gru_gemv_kernel(const int* __restrict__ tok,
                                const float* __restrict__ hidden,
                                const float* __restrict__ emb,
                                const float* __restrict__ W_ih,
                                const float* __restrict__ W_hh,
                                const float* __restrict__ b_ih,
                                const float* __restrict__ b_hh,
                                float* __restrict__ gi,
                                float* __restrict__ gh)
{
    const int tid  = blockIdx.x * blockDim.x + threadIdx.x;
    const int w    = tid >> 5;            // global wave id, [0, 6144)
    const int lane = tid & 31;
    const bool is_ih = (w < H3);          // wave-uniform
    const int row  = is_ih ? w : (w - H3);

    const float* Wr = (is_ih ? W_ih : W_hh) + (size_t)row * Hdim;
    const float4* W4 = (const float4*)Wr;
    const int idx = tok[0];
    const float4* v4 = (const float4*)(is_ih ? (emb + (size_t)idx * Hdim)
                                             : hidden);
    float acc = 0.0f;
#pragma unroll
    for (int j = 0; j < 8; ++j) {
        const int e = j * 32 + lane;      // coalesced across lanes
        float4 wv = W4[e];
        float4 xv = v4[e];
        if (is_ih) {                      // x = relu(emb[tok])
            xv.x = fmaxf(xv.x, 0.0f); xv.y = fmaxf(xv.y, 0.0f);
            xv.z = fmaxf(xv.z, 0.0f); xv.w = fmaxf(xv.w, 0.0f);
        }
        acc = fmaf(wv.x, xv.x, acc);
        acc = fmaf(wv.y, xv.y, acc);
        acc = fmaf(wv.z, xv.z, acc);
        acc = fmaf(wv.w, xv.w, acc);
    }
#pragma unroll
    for (int off = 16; off > 0; off >>= 1)
        acc += __shfl_xor(acc, off, 32);

    if (lane == 0) {
        if (is_ih) gi[row] = acc + b_ih[row];
        else       gh[row] = acc + b_hh[row];
    }
}

// ---------------------------------------------------------------------------
// Kernel 2: GRU gate combine (PyTorch gate order r,z,n) -> h_new
// ---------------------------------------------------------------------------
__global__ void gru_combine_kernel(const float* __restrict__ gi,
                                   const float* __restrict__ gh,
                                   const float* __restrict__ hidden,
                                   float* __restrict__ hnew)
{
    const int j = blockIdx.x * blockDim.x + threadIdx.x;
    if (j >= Hdim) return;
    const float r = 1.0f / (1.0f + expf(-(gi[j]          + gh[j])));
    const float z = 1.0f / (1.0f + expf(-(gi[Hdim + j]   + gh[Hdim + j])));
    const float n = tanhf(gi[2 * Hdim + j] + r * gh[2 * Hdim + j]);
    hnew[j] = (1.0f - z) * n + z * hidden[j];
}

// ---------------------------------------------------------------------------
// Kernel 3: logits[v] = W_out[v,:]·h_new + b_out[v] via V_WMMA_F32_16X16X4_F32
//
// Wave owns 16 rows. Per K-chunk of 8:
//   lane l loads float4 W_out[base + l%16, k0 + (l>>4)*4 .. +3]     (b128)
//   h broadcast float4 from LDS: sh4[t*2 + (l>>4)]
// K-slot remap (A/B agree, so the contraction is exact):
//   WMMA#1: hwK = {k0,k0+1 | k0+4,k0+5}  -> A = {f4.x,f4.y}, B = {h.x,h.y}
//   WMMA#2: hwK = {k0+2,k0+3 | k0+6,k0+7}-> A = {f4.z,f4.w}, B = {h.z,h.w}
// B columns all replicate h, so every D column holds the 16 dot products.
// ---------------------------------------------------------------------------
__global__ void logits_wmma_kernel(const float* __restrict__ W_out,
                                   const float* __restrict__ b_out,
                                   const float* __restrict__ hnew,
                                   float* __restrict__ logits)
{
    __shared__ float4 sh4[Hdim / 4];                 // 4KB: h_new staged once
    const int tid = threadIdx.x;
    sh4[tid] = ((const float4*)hnew)[tid];
    __syncthreads();

    const int wave = tid >> 5;
    const int lane = tid & 31;
    const int g = blockIdx.x * 8 + wave;             // 16-row group id
    if (g >= NGROUPS) return;                        // after the barrier

    const int base = g * 16;
    int rowL = base + (lane & 15);
    if (rowL >= Vdim) rowL = Vdim - 1;               // clamp loads, mask stores
    const int half = lane >> 4;                      // 0: lanes 0-15, 1: 16-31

    const float4* wrow = (const float4*)(W_out + (size_t)rowL * Hdim) + half;

    v8f c = {0.f, 0.f, 0.f, 0.f, 0.f, 0.f, 0.f, 0.f};

    for (int t = 0; t < Hdim / 8; ++t) {             // 128 iters, K step 8
        const float4 wf = wrow[t * 2];
        __builtin_prefetch(&wrow[t * 2 + 32], 0, 0); // global_prefetch_b8
        const float4 hb = sh4[t * 2 + half];         // LDS 2-way broadcast

        v2f a1 = {wf.x, wf.y}, b1 = {hb.x, hb.y};
        v2f a2 = {wf.z, wf.w}, b2 = {hb.z, hb.w};
        c = __builtin_amdgcn_wmma_f32_16x16x4_f32(false, a1, false, b1,
                                                  (short)0, c, false, false);
        c = __builtin_amdgcn_wmma_f32_16x16x4_f32(false, a2, false, b2,
                                                  (short)0, c, false, false);
    }

    // D layout: VGPR v, lanes 0-15 -> (M=v, N=lane); lanes 16-31 -> (M=v+8).
    // Column 0 (lanes 0 and 16) carries all 16 row results.
    if (lane == 0) {
#pragma unroll
        for (int v = 0; v < 8; ++v) {
            const int r = base + v;
            if (r < Vdim) logits[r] = c[v] + b_out[r];
        }
    } else if (lane == 16) {
#pragma unroll
        for (int v = 0; v < 8; ++v) {
            const int r = base + 8 + v;
            if (r < Vdim) logits[r] = c[v] + b_out[r];
        }
    }
}

// ---------------------------------------------------------------------------
// Kernels 4-6: numerically stable log-softmax (online max/sum-exp).
// ---------------------------------------------------------------------------
__global__ void lse_partial_kernel(const float* __restrict__ logits,
                                   float* __restrict__ pmax,
                                   float* __restrict__ psum)
{
    __shared__ float sm[256];
    __shared__ float ss[256];
    const int tid = threadIdx.x;
    float m = -3.4e38f, s = 0.0f;
    for (int i = blockIdx.x * 256 + tid; i < Vdim; i += 64 * 256) {
        const float x = logits[i];
        if (x > m) { s = s * expf(m - x) + 1.0f; m = x; }
        else       { s += expf(x - m); }
    }
    sm[tid] = m; ss[tid] = s;
    __syncthreads();
    for (int st = 128; st > 0; st >>= 1) {
        if (tid < st) {
            const float m1 = sm[tid], s1 = ss[tid];
            const float m2 = sm[tid + st], s2 = ss[tid + st];
            const float M = fmaxf(m1, m2);
            sm[tid] = M;
            ss[tid] = s1 * expf(m1 - M) + s2 * expf(m2 - M);
        }
        __syncthreads();
    }
    if (tid == 0) { pmax[blockIdx.x] = sm[0]; psum[blockIdx.x] = ss[0]; }
}

__global__ void lse_final_kernel(const float* __restrict__ pmax,
                                 const float* __restrict__ psum,
                                 float* __restrict__ stats)
{
    if (threadIdx.x == 0) {
        float m = -3.4e38f, s = 0.0f;
        for (int i = 0; i < 64; ++i) {
            const float m2 = pmax[i], s2 = psum[i];
            const float M = fmaxf(m, m2);
            s = s * expf(m - M) + s2 * expf(m2 - M);
            m = M;
        }
        stats[0] = m;
        stats[1] = logf(s);
    }
}

__global__ void lsm_write_kernel(float* __restrict__ logits,
                                 const float* __restrict__ stats)
{
    const int i = blockIdx.x * blockDim.x + threadIdx.x;
    if (i < Vdim) logits[i] = logits[i] - stats[0] - stats[1];
}

// ---------------------------------------------------------------------------
// Launch: gi/gh live temporarily in d_out's logits region (dead before the
// logits kernel overwrites it); h_new lands directly in its output slot.
// ---------------------------------------------------------------------------
extern "C" void kernel_launch(void* const* d_in, const int* in_sizes, int n_in,
                              void* d_out, int out_size, void* d_ws, size_t ws_size,
                              hipStream_t stream)
{
    const int*   tok    = (const int*)  d_in[0];
    const float* hidden = (const float*)d_in[1];
    const float* emb    = (const float*)d_in[2];
    const float* W_ih   = (const float*)d_in[3];
    const float* W_hh   = (const float*)d_in[4];
    const float* b_ih   = (const float*)d_in[5];
    const float* b_hh   = (const float*)d_in[6];
    const float* W_out  = (const float*)d_in[7];
    const float* b_out  = (const float*)d_in[8];

    float* out   = (float*)d_out;
    float* gi    = out;              // scratch in logits region (overwritten)
    float* gh    = out + H3;
    float* hnew  = out + Vdim;       // final output slot, input to projection
    float* wsF   = (float*)d_ws;
    float* pmax  = wsF;
    float* psum  = wsF + 64;
    float* stats = wsF + 128;

    gru_gemv_kernel<<<768, 256, 0, stream>>>(tok, hidden, emb, W_ih, W_hh,
                                             b_ih, b_hh, gi, gh);
    gru_combine_kernel<<<4, 256, 0, stream>>>(gi, gh, hidden, hnew);

    const int cblocks = (NGROUPS + 7) / 8;           // 8 waves per block
    logits_wmma_kernel<<<cblocks, 256, 0, stream>>>(W_out, b_out, hnew, out);

    lse_partial_kernel<<<64, 256, 0, stream>>>(out, pmax, psum);
    lse_final_kernel<<<1, 32, 0, stream>>>(pmax, psum, stats);
    lsm_write_kernel<<<(Vdim + 255) / 256, 256, 0, stream>>>(out, stats);
}